// HierarchicalGNN_22033182228984
// MI455X (gfx1250) — compile-verified
//
#include <hip/hip_runtime.h>
#include <hip/hip_bf16.h>

typedef __attribute__((ext_vector_type(2))) float v2f;
typedef __attribute__((ext_vector_type(8))) float v8f;

#define NGRAPHS 64

// ---------------------------------------------------------------- utilities
__global__ void fill_kernel(float* __restrict__ p, float v, int n) {
  int i = blockIdx.x * blockDim.x + threadIdx.x;
  if (i < n) p[i] = v;
}

__global__ void count_deg_kernel(const int* __restrict__ dst, float* __restrict__ deg, int E) {
  int e = blockIdx.x * blockDim.x + threadIdx.x;
  if (e < E)
    __hip_atomic_fetch_add(&deg[dst[e]], 1.0f, __ATOMIC_RELAXED, __HIP_MEMORY_SCOPE_AGENT);
}

__global__ void dinv_kernel(const float* __restrict__ deg, float* __restrict__ dinv, int N) {
  int i = blockIdx.x * blockDim.x + threadIdx.x;
  if (i < N) dinv[i] = rsqrtf(deg[i]);   // deg >= 1 always (self loops)
}

// ---------------------------------------------------------------- GEMM (f32 WMMA 16x16x4)
// C[N x NC] = act(A[N x 128]) @ W[128 x NC]
// If bias != nullptr, the A operand is relu(A + bias[k]) (fused layer-1 activation).
template <int NC>
__global__ __launch_bounds__((NC / 16) * 32)
void gemm_wmma_kernel(const float* __restrict__ A, const float* __restrict__ W,
                      float* __restrict__ C, const float* __restrict__ bias, int N) {
  __shared__ float As[16][132];   // 132*4 bytes per row: 16B aligned, conflict-free
  const int K = 128;
  const int rowbase = blockIdx.x * 16;
  const int nthr = (NC / 16) * 32;

  // cooperative stage of a 16x128 strip of A into LDS (with optional relu+bias)
  for (int idx = threadIdx.x * 4; idx < 16 * K; idx += nthr * 4) {
    int rr = idx >> 7;          // /128
    int cc = idx & 127;
    int grow = rowbase + rr;
    if (grow >= N) grow = N - 1;   // clamp (stores guarded below)
    float4 f = *(const float4*)(A + (size_t)grow * K + cc);
    if (bias != nullptr) {
      f.x = fmaxf(f.x + bias[cc + 0], 0.0f);
      f.y = fmaxf(f.y + bias[cc + 1], 0.0f);
      f.z = fmaxf(f.z + bias[cc + 2], 0.0f);
      f.w = fmaxf(f.w + bias[cc + 3], 0.0f);
    }
    As[rr][cc + 0] = f.x;
    As[rr][cc + 1] = f.y;
    As[rr][cc + 2] = f.z;
    As[rr][cc + 3] = f.w;
  }
  __syncthreads();

  const int wave = threadIdx.x >> 5;     // one 16-col tile per wave
  const int lane = threadIdx.x & 31;
  const int half = lane >> 4;            // 0: lanes 0-15, 1: lanes 16-31
  const int l16  = lane & 15;
  const int col  = wave * 16 + l16;
  const int kb   = half * 2;             // A frag: K=0,1 (half 0) / K=2,3 (half 1)
  const float* wp = W + (size_t)kb * NC + col;   // rows kb, kb+1 of each K-quad

  v8f acc = {};
#pragma unroll 8
  for (int k0 = 0; k0 < K; k0 += 4) {
    v2f a, b;
    a.x = As[l16][k0 + kb];
    a.y = As[l16][k0 + kb + 1];
    b.x = wp[(size_t)k0 * NC];
    b.y = wp[(size_t)k0 * NC + NC];
    acc = __builtin_amdgcn_wmma_f32_16x16x4_f32(false, a, false, b, (short)0, acc,
                                                false, false);
  }

  // C/D layout: VGPR i -> row rowoff+i, col = lane&15.  Row bounds are wave-uniform,
  // so take one uniform branch: full tiles store branchlessly (immediate offsets).
  const int rowoff = half * 8;
  float* cp = C + (size_t)(rowbase + rowoff) * NC + col;
  if (rowbase + 16 <= N) {
#pragma unroll
    for (int i = 0; i < 8; ++i)
      cp[(size_t)i * NC] = acc[i];
  } else {
#pragma unroll
    for (int i = 0; i < 8; ++i)
      if (rowbase + rowoff + i < N) cp[(size_t)i * NC] = acc[i];
  }
}

// ---------------------------------------------------------------- self-loop init
// agg[i, :] = dinv[i]^2 * xw[i, :]   (initializes the accumulator, no memset needed)
template <int D>
__global__ void selfloop_kernel(const float* __restrict__ xw, const float* __restrict__ dinv,
                                float* __restrict__ agg, int N) {
  const int DV = D / 4;
  int idx = blockIdx.x * blockDim.x + threadIdx.x;
  if (idx >= N * DV) return;
  int node = idx / DV;
  float s = dinv[node];
  s = s * s;
  float4 f = ((const float4*)xw)[idx];
  float4 o;
  o.x = f.x * s; o.y = f.y * s; o.z = f.z * s; o.w = f.w * s;
  ((float4*)agg)[idx] = o;
}

// ---------------------------------------------------------------- edge scatter-add
// one wave per edge; lanes cover the feature dim (VEC floats each).
// agg[dst, :] += dinv[src]*dinv[dst] * xw[src, :]
template <int VEC>   // D = 32*VEC
__global__ void edge_agg_kernel(const int* __restrict__ src, const int* __restrict__ dst,
                                const float* __restrict__ dinv, const float* __restrict__ xw,
                                float* __restrict__ agg, int E) {
  int gid = blockIdx.x * blockDim.x + threadIdx.x;
  int e = gid >> 5;
  if (e >= E) return;
  int lane = gid & 31;
  int r = src[e];
  int c = dst[e];
  float nrm = dinv[r] * dinv[c];
  const int D = 32 * VEC;
  const float* xp = xw + (size_t)r * D + lane * VEC;
  float* ap = agg + (size_t)c * D + lane * VEC;
#pragma unroll
  for (int v = 0; v < VEC; ++v) {
    __hip_atomic_fetch_add(ap + v, nrm * xp[v], __ATOMIC_RELAXED, __HIP_MEMORY_SCOPE_AGENT);
  }
}

// ---------------------------------------------------------------- pooling
__global__ void cnt_kernel(const int* __restrict__ batch, float* __restrict__ cnt, int N) {
  int i = blockIdx.x * blockDim.x + threadIdx.x;
  if (i < N)
    __hip_atomic_fetch_add(&cnt[batch[i]], 1.0f, __ATOMIC_RELAXED, __HIP_MEMORY_SCOPE_AGENT);
}

__global__ void pool_sum_kernel(const int* __restrict__ batch, const float* __restrict__ agg2,
                                float* __restrict__ sums, int N) {
  int idx = blockIdx.x * blockDim.x + threadIdx.x;   // N*64
  if (idx >= N * 64) return;
  int i = idx >> 6;
  int d = idx & 63;
  __hip_atomic_fetch_add(&sums[batch[i] * 64 + d], agg2[idx],
                         __ATOMIC_RELAXED, __HIP_MEMORY_SCOPE_AGENT);
}

__global__ void finalize_kernel(const float* __restrict__ sums, const float* __restrict__ cnt,
                                const float* __restrict__ b2, float* __restrict__ out) {
  int idx = blockIdx.x * blockDim.x + threadIdx.x;   // 64*64
  if (idx >= NGRAPHS * 64) return;
  int g = idx >> 6;
  int d = idx & 63;
  out[idx] = sums[idx] / fmaxf(cnt[g], 1.0f) + b2[d];
}

// ---------------------------------------------------------------- launch
extern "C" void kernel_launch(void* const* d_in, const int* in_sizes, int n_in,
                              void* d_out, int out_size, void* d_ws, size_t ws_size,
                              hipStream_t stream) {
  const float* x     = (const float*)d_in[0];   // [N,128]
  const int*   ei    = (const int*)d_in[1];     // [2,E]
  const int*   batch = (const int*)d_in[2];     // [N]
  const float* W1    = (const float*)d_in[3];   // [128,128]
  const float* b1    = (const float*)d_in[4];   // [128]
  const float* W2    = (const float*)d_in[5];   // [128,64]
  const float* b2    = (const float*)d_in[6];   // [64]
  float* out = (float*)d_out;

  const int N = in_sizes[2];
  const int E = in_sizes[1] / 2;
  const int* src = ei;
  const int* dst = ei + E;

  // workspace layout (floats)
  float* ws = (float*)d_ws;
  size_t o = 0;
  float* xw1  = ws + o; o += (size_t)N * 128;   // layer1 transform; region reused below
  float* agg1 = ws + o; o += (size_t)N * 128;   // layer1 aggregate (H after fused relu+b1)
  float* xw2  = xw1;                            // reuse: first N*64 of xw1 region
  float* agg2 = xw1 + (size_t)N * 64;           // reuse: second N*64 of xw1 region
  float* deg  = ws + o; o += (size_t)N;
  float* dinv = ws + o; o += (size_t)N;
  float* sums = ws + o; o += NGRAPHS * 64;      // sums then cnt contiguous -> one zero fill
  float* cnt  = ws + o; o += NGRAPHS;

  const int T = 256;
  // degrees / symmetric norm (shared by both layers)
  fill_kernel<<<(N + T - 1) / T, T, 0, stream>>>(deg, 1.0f, N);               // self loop
  count_deg_kernel<<<(E + T - 1) / T, T, 0, stream>>>(dst, deg, E);
  dinv_kernel<<<(N + T - 1) / T, T, 0, stream>>>(deg, dinv, N);

  // layer 1: xw1 = x @ W1 ; agg1 = scatter(norm * xw1) (+self loop)
  gemm_wmma_kernel<128><<<(N + 15) / 16, 256, 0, stream>>>(x, W1, xw1, nullptr, N);
  selfloop_kernel<128><<<((N * 32) + T - 1) / T, T, 0, stream>>>(xw1, dinv, agg1, N);
  edge_agg_kernel<4><<<((size_t)E * 32 + T - 1) / T, T, 0, stream>>>(src, dst, dinv, xw1, agg1, E);

  // layer 2: xw2 = relu(agg1 + b1) @ W2 ; agg2 = scatter(norm * xw2) (+self loop)
  gemm_wmma_kernel<64><<<(N + 15) / 16, 128, 0, stream>>>(agg1, W2, xw2, b1, N);
  selfloop_kernel<64><<<((N * 16) + T - 1) / T, T, 0, stream>>>(xw2, dinv, agg2, N);
  edge_agg_kernel<2><<<((size_t)E * 32 + T - 1) / T, T, 0, stream>>>(src, dst, dinv, xw2, agg2, E);

  // global mean pool (+b2 folded into finalize)
  fill_kernel<<<(NGRAPHS * 64 + NGRAPHS + T - 1) / T, T, 0, stream>>>(sums, 0.0f,
                                                                      NGRAPHS * 64 + NGRAPHS);
  cnt_kernel<<<(N + T - 1) / T, T, 0, stream>>>(batch, cnt, N);
  pool_sum_kernel<<<((size_t)N * 64 + T - 1) / T, T, 0, stream>>>(batch, agg2, sums, N);
  finalize_kernel<<<(NGRAPHS * 64 + T - 1) / T, T, 0, stream>>>(sums, cnt, b2, out);
}